// LTCCell_326417514660
// MI455X (gfx1250) — compile-verified
//
#include <hip/hip_runtime.h>
#include <cstdint>

// ---------------------------------------------------------------------------
// LTC cell for MI455X (gfx1250).
// Roofline: ~436M sigmoid-elements (exp2+rcp each) vs ~7MB HBM traffic ->
// transcendental-pipe bound. Strategy: fold params once, park them in the
// 320KB/WGP LDS (staged with gfx1250 async-to-LDS ops, overlapped with the
// sensory phase), lane=unit so param reads are stride-1 conflict-free b128
// LDS loads, state read as float4 broadcasts. Output affine on the WMMA pipe
// (exact: diagonal B matrix, f32 16x16x4).
// ---------------------------------------------------------------------------

#define B_TOT   4096
#define SENS    64
#define UNITS   128
#define MOTOR   32
#define UNFOLDS 6
#define LOG2E_F 1.4426950408889634f
#define EPS_F   1e-8f

typedef __attribute__((ext_vector_type(2))) float v2f;
typedef __attribute__((ext_vector_type(8))) float v8f;
typedef int v4i_vs __attribute__((vector_size(16)));   // b128 payload type

__device__ __forceinline__ float dexp2(float x) {
#if defined(__HIP_DEVICE_COMPILE__)
  return __builtin_amdgcn_exp2f(x);   // v_exp_f32
#else
  return x;
#endif
}
__device__ __forceinline__ float drcp(float x) {
#if defined(__HIP_DEVICE_COMPILE__)
  return __builtin_amdgcn_rcpf(x);    // v_rcp_f32
#else
  return x;
#endif
}

// gfx1250 async global->LDS staging (ASYNCcnt path). Guarded: falls back to
// a plain VGPR round-trip copy if the toolchain lacks the builtins.
#if defined(__HIP_DEVICE_COMPILE__) && \
    __has_builtin(__builtin_amdgcn_global_load_async_to_lds_b128) && \
    __has_builtin(__builtin_amdgcn_s_wait_asynccnt)
#define HAVE_ASYNC_LDS 1
#endif

#if defined(HAVE_ASYNC_LDS)
__device__ __forceinline__ void async_cp16(const void* g, void* l) {
  // Builtin signature (from probe): (int4 addrspace(1)*, int4 addrspace(3)*,
  // imm offset, imm cpol). Lowers to global_load_async_to_lds_b128.
  __builtin_amdgcn_global_load_async_to_lds_b128(
      (__attribute__((address_space(1))) v4i_vs*)(uintptr_t)g,
      (__attribute__((address_space(3))) v4i_vs*)(uint32_t)(uintptr_t)l, 0, 0);
}
#endif

// ---------------------------------------------------------------------------
// Kernel 1: fold parameters.
//   Rp[i*128+u] = { -sigma*log2e, sigma*mu*log2e, w*mask*erev, w*mask }
//   Sp[s*128+u] = same folding with sensory params.
// sigmoid(sigma*(v-mu)) == 1/(1+exp2(fma(-sigma*log2e, v, sigma*mu*log2e)))
// NOTE: jax bool arrays are 1 byte/elem -> masks read as unsigned char.
// ---------------------------------------------------------------------------
__global__ __launch_bounds__(256) void ltc_pack(
    const float* __restrict__ sigma, const float* __restrict__ mu,
    const float* __restrict__ w,     const float* __restrict__ erev,
    const unsigned char* __restrict__ mask,
    const float* __restrict__ ssig,  const float* __restrict__ smu,
    const float* __restrict__ sw,    const float* __restrict__ serev,
    const unsigned char* __restrict__ smask,
    float4* __restrict__ Rp, float4* __restrict__ Sp) {
  int idx = blockIdx.x * 256 + threadIdx.x;
  if (idx < UNITS * UNITS) {
    float sg = sigma[idx], m = mu[idx];
    float ww = mask[idx] ? w[idx] : 0.0f;
    Rp[idx] = make_float4(-sg * LOG2E_F, sg * m * LOG2E_F, ww * erev[idx], ww);
  }
  if (idx < SENS * UNITS) {
    float sg = ssig[idx], m = smu[idx];
    float ww = smask[idx] ? sw[idx] : 0.0f;
    Sp[idx] = make_float4(-sg * LOG2E_F, sg * m * LOG2E_F, ww * serev[idx], ww);
  }
}

// ---------------------------------------------------------------------------
// Kernel 2: main ODE solver. 512 threads = 16 wave32s. Block owns 4 batches.
// wave w: batch = w/4, u-chunk = w%4 (32 units per wave, lane = unit).
// LDS: P[16384] float4 (sensory parked in upper half, recurrent fills all),
// v double buffer, x. 1024 blocks -> saturates any plausible WGP count.
// ---------------------------------------------------------------------------
__global__ __launch_bounds__(512, 1) void ltc_main(
    const float* __restrict__ inputs, const float* __restrict__ state,
    const float* __restrict__ gleak,  const float* __restrict__ vleak,
    const float* __restrict__ cm,
    const float* __restrict__ input_w, const float* __restrict__ input_b,
    const float4* __restrict__ Rp, const float4* __restrict__ Sp,
    float* __restrict__ next_state) {
  extern __shared__ __align__(16) char smem[];
  float4* P    = (float4*)smem;                               // 16384 float4 = 256KB
  float*  vbuf = (float*)(smem + 16384 * sizeof(float4));     // [2][4][128] = 4KB
  float*  xs   = (float*)(smem + 16384 * sizeof(float4) + 2 * 4 * UNITS * sizeof(float)); // [4][64]

  const int tid  = threadIdx.x;
  const int lane = tid & 31;
  const int wv   = tid >> 5;      // 0..15
  const int b_l  = wv >> 2;       // 0..3 batch-in-block
  const int uc   = wv & 3;        // 0..3 unit chunk
  const int u    = uc * 32 + lane;
  const int b0   = blockIdx.x * 4;
  const int b    = b0 + b_l;

  // Stage x = inputs*input_w + input_b for this block's 4 batches.
  if (tid < 4 * SENS) {
    int bb = tid >> 6, s = tid & 63;
    xs[bb * SENS + s] = fmaf(inputs[(b0 + bb) * SENS + s], input_w[s], input_b[s]);
  }
  // Stage initial state into buffer 0.
  {
    int bb = tid >> 7, uu = tid & 127;
    vbuf[bb * UNITS + uu] = state[(b0 + bb) * UNITS + uu];
  }

#if defined(HAVE_ASYNC_LDS)
  // Async staging, overlapped with the sensory phase:
  //   1) sensory params -> P[8192..16383]   (needed first)
  //   2) recurrent lower half -> P[0..8191] (hidden behind sensory compute)
  // Async loads complete in order per wave, so waiting for <=16 outstanding
  // releases exactly the 16 sensory transfers.
  for (int k = 0; k < 16; ++k)
    async_cp16(Sp + k * 512 + tid, P + 8192 + k * 512 + tid);
  for (int k = 0; k < 16; ++k)
    async_cp16(Rp + k * 512 + tid, P + k * 512 + tid);
  __builtin_amdgcn_s_wait_asynccnt(16);   // sensory half landed
#else
  for (int k = 0; k < 16; ++k)
    P[8192 + k * 512 + tid] = Sp[k * 512 + tid];
  // Warm the recurrent block in cache during the sensory phase.
  __builtin_prefetch((const char*)Rp + tid * 512, 0, 0);
#endif
  __syncthreads();

  // Sensory accumulation (once): per-lane (b_l, u), reduce over s.
  float sN = 0.0f, sD = 0.0f;
  {
    const float4* xv4 = (const float4*)(xs + b_l * SENS);
#pragma unroll 4
    for (int s4 = 0; s4 < SENS / 4; ++s4) {
      float4 xq = xv4[s4];            // wave-uniform b128 broadcast (4 x values)
      const float xr[4] = {xq.x, xq.y, xq.z, xq.w};
#pragma unroll
      for (int j = 0; j < 4; ++j) {
        float4 p = P[8192 + (s4 * 4 + j) * UNITS + u];  // per-lane b128, stride-1
        float  e = dexp2(fmaf(p.x, xr[j], p.y));
        float  g = drcp(1.0f + e);    // sigmoid
        sN = fmaf(p.z, g, sN);
        sD = fmaf(p.w, g, sD);
      }
    }
  }
  __syncthreads();  // all waves done reading sensory params

#if defined(HAVE_ASYNC_LDS)
  // 3) recurrent upper half -> P[8192..16383] (overwrites sensory params)
  for (int k = 0; k < 16; ++k)
    async_cp16(Rp + 8192 + k * 512 + tid, P + 8192 + k * 512 + tid);
  __builtin_amdgcn_s_wait_asynccnt(0);
#else
  for (int k = 0; k < 32; ++k)
    P[k * 512 + tid] = Rp[k * 512 + tid];
#endif
  __syncthreads();

  const float glk = gleak[u];
  const float gv  = glk * vleak[u];
  const float cmt = cm[u] * (float)UNFOLDS;  // cm / (elapsed/unfolds)

  int cur = 0;
  for (int step = 0; step < UNFOLDS; ++step) {
    const float* vc = vbuf + cur * (4 * UNITS) + b_l * UNITS;
    const float4* vc4 = (const float4*)vc;   // 16B-aligned (2KB offsets)
    float wn = sN, wd = sD;
#pragma unroll 4
    for (int i4 = 0; i4 < UNITS / 4; ++i4) {
      float4 vq = vc4[i4];            // one b128 broadcast serves 4 elements
      const float vr[4] = {vq.x, vq.y, vq.z, vq.w};
#pragma unroll
      for (int j = 0; j < 4; ++j) {
        float4 p = P[(i4 * 4 + j) * UNITS + u];
        float  e = dexp2(fmaf(p.x, vr[j], p.y));
        float  g = drcp(1.0f + e);
        wn = fmaf(p.z, g, wn);
        wd = fmaf(p.w, g, wd);
      }
    }
    float vpre = vc[u];
    float num  = fmaf(cmt, vpre, gv) + wn;
    float den  = cmt + glk + wd + EPS_F;
    float vnew = num / den;           // IEEE div, once per 128 sigmoids
    vbuf[(cur ^ 1) * (4 * UNITS) + b_l * UNITS + u] = vnew;
    __syncthreads();
    cur ^= 1;
  }

  next_state[b * UNITS + u] = vbuf[cur * (4 * UNITS) + b_l * UNITS + u];
}

// ---------------------------------------------------------------------------
// Kernel 3: output affine on the WMMA pipe.
// out[b, n] = state[b, n] * output_w[n] + output_b[n], n < 32.
// D = A x B + C with B = diag(output_w) (exact in f32). One wave owns a
// 16(b) x 16(n) tile; 4x V_WMMA_F32_16X16X4_F32 walk the diagonal K range.
// output_w/output_b are loaded ONCE into registers so the diagonal build is
// pure v_cndmask (no divergent loads).
// ---------------------------------------------------------------------------
__global__ __launch_bounds__(256) void ltc_out(
    const float* __restrict__ st, const float* __restrict__ ow,
    const float* __restrict__ ob, float* __restrict__ out) {
  const int wvg   = (blockIdx.x * blockDim.x + threadIdx.x) >> 5;  // 0..511
  const int lane  = threadIdx.x & 31;
  const int ntile = wvg & 1;        // n0 = 0 or 16
  const int btile = wvg >> 1;       // 0..255
  const int n0    = ntile * 16;
  const int mrow  = lane & 15;      // A-matrix M = lane%16 (both halves)
  const int hi    = lane >> 4;      // half-wave select
  const int nn    = n0 + (lane & 15);

  const float wnn  = ow[nn];        // hoisted: registers, not per-chunk loads
  const float bias = ob[nn];

#if defined(__HIP_DEVICE_COMPILE__) && __has_builtin(__builtin_amdgcn_wmma_f32_16x16x4_f32)
  v8f c;
#pragma unroll
  for (int r = 0; r < 8; ++r) c[r] = bias;
#pragma unroll
  for (int kc = 0; kc < 4; ++kc) {
    const int k0  = n0 + kc * 4;          // diagonal K chunk
    const int kk0 = k0 + hi * 2;          // A VGPR0/1 hold K = kk0, kk0+1
    v2f a, bm;
    a.x  = st[(btile * 16 + mrow) * UNITS + kk0];
    a.y  = st[(btile * 16 + mrow) * UNITS + kk0 + 1];
    bm.x = (kk0 == nn)     ? wnn : 0.0f;  // v_cndmask, branch-free
    bm.y = (kk0 + 1 == nn) ? wnn : 0.0f;
    c = __builtin_amdgcn_wmma_f32_16x16x4_f32(false, a, false, bm,
                                              (short)0, c, false, false);
  }
#pragma unroll
  for (int r = 0; r < 8; ++r) {
    const int m = r + hi * 8;             // C/D layout: VGPR r -> M=r / r+8
    out[(btile * 16 + m) * MOTOR + nn] = c[r];
  }
#else
  // Scalar fallback, identical element mapping.
  for (int r = 0; r < 8; ++r) {
    const int m  = r + hi * 8;
    const int bb = btile * 16 + m;
    out[bb * MOTOR + nn] = fmaf(st[bb * UNITS + nn], wnn, bias);
  }
#endif
}

// ---------------------------------------------------------------------------
extern "C" void kernel_launch(void* const* d_in, const int* in_sizes, int n_in,
                              void* d_out, int out_size, void* d_ws, size_t ws_size,
                              hipStream_t stream) {
  (void)in_sizes; (void)n_in; (void)out_size; (void)ws_size;

  const float* inputs  = (const float*)d_in[0];
  const float* state   = (const float*)d_in[1];
  const float* gleak   = (const float*)d_in[2];
  const float* vleak   = (const float*)d_in[3];
  const float* cm      = (const float*)d_in[4];
  const float* sigma   = (const float*)d_in[5];
  const float* mu      = (const float*)d_in[6];
  const float* w       = (const float*)d_in[7];
  const float* erev    = (const float*)d_in[8];
  const float* ssig    = (const float*)d_in[9];
  const float* smu     = (const float*)d_in[10];
  const float* sw      = (const float*)d_in[11];
  const float* serev   = (const float*)d_in[12];
  const float* iw      = (const float*)d_in[13];
  const float* ibias   = (const float*)d_in[14];
  const float* ow      = (const float*)d_in[15];
  const float* obias   = (const float*)d_in[16];
  const unsigned char* mask  = (const unsigned char*)d_in[17];  // jax bool = 1B
  const unsigned char* smask = (const unsigned char*)d_in[18];

  // Workspace: folded params. 256KB recurrent + 128KB sensory.
  float4* Rp = (float4*)d_ws;
  float4* Sp = Rp + UNITS * UNITS;

  // d_out = [out (4096x32) | next_state (4096x128)] flat in return order.
  float* out = (float*)d_out;
  float* nst = out + (size_t)B_TOT * MOTOR;

  // 1) fold params (covers both 16384 recurrent and 8192 sensory entries)
  ltc_pack<<<(UNITS * UNITS + 255) / 256, 256, 0, stream>>>(
      sigma, mu, w, erev, mask, ssig, smu, sw, serev, smask, Rp, Sp);

  // 2) ODE solve: 261KB dynamic LDS (legal: <=320KB per workgroup on CDNA5)
  const size_t lds_bytes = 16384 * sizeof(float4)
                         + 2 * 4 * UNITS * sizeof(float)
                         + 4 * SENS * sizeof(float);
  (void)hipFuncSetAttribute((const void*)ltc_main,
                            hipFuncAttributeMaxDynamicSharedMemorySize,
                            (int)lds_bytes);
  ltc_main<<<B_TOT / 4, 512, lds_bytes, stream>>>(
      inputs, state, gleak, vleak, cm, iw, ibias, Rp, Sp, nst);

  // 3) output affine on the matrix pipe: 512 waves (256 b-tiles x 2 n-tiles)
  ltc_out<<<(512 * 32) / 256, 256, 0, stream>>>(nst, ow, obias, out);
}